// LiteLocalProjection3D_86912958202324
// MI455X (gfx1250) — compile-verified
//
#include <hip/hip_runtime.h>
#include <math.h>

#define NSP 110592   // 48^3 downsampled voxels
#define NIN 884736   // 96^3
#define G_CNT 1728   // 12^3 centers
#define W3 343       // 7^3 window

typedef float v2f __attribute__((ext_vector_type(2)));
typedef float v8f __attribute__((ext_vector_type(8)));

// ---------------- Kernel 1: depthwise 3x3x3 stride-2 conv + bias ----------------
__global__ __launch_bounds__(256) void dwconv_kernel(const float* __restrict__ feat,
        const float* __restrict__ dw_w, const float* __restrict__ dw_b,
        float* __restrict__ g)
{
    int idx = blockIdx.x * 256 + threadIdx.x;
    int c  = idx / NSP;
    int n  = idx - c * NSP;
    int od = n / (48 * 48);
    int r  = n - od * 48 * 48;
    int oh = r / 48;
    int ow = r - oh * 48;
    const float* in = feat + (size_t)c * NIN;
    const float* w  = dw_w + c * 27;
    float acc = dw_b[c];
    int iz0 = 2 * od - 1, iy0 = 2 * oh - 1, ix0 = 2 * ow - 1;
#pragma unroll
    for (int kz = 0; kz < 3; ++kz) {
        int iz = iz0 + kz;
        if (iz < 0 || iz >= 96) continue;
#pragma unroll
        for (int ky = 0; ky < 3; ++ky) {
            int iy = iy0 + ky;
            if (iy < 0 || iy >= 96) continue;
            const float* row = in + ((size_t)iz * 96 + iy) * 96;
            const float* wr  = w + kz * 9 + ky * 3;
#pragma unroll
            for (int kx = 0; kx < 3; ++kx) {
                int ix = ix0 + kx;
                if (ix >= 0 && ix < 96) acc = fmaf(row[ix], wr[kx], acc);
            }
        }
    }
    g[idx] = acc;
}

// ---------------- Kernel 2: pointwise f = relu(pw_w(32x64) @ g + b), fp32 WMMA ----------------
__global__ __launch_bounds__(64) void gemm1_kernel(const float* __restrict__ g,
        const float* __restrict__ pw_w, const float* __restrict__ pw_b,
        float* __restrict__ f)
{
    int lane = threadIdx.x & 31;
    int wv   = threadIdx.x >> 5;            // row-tile 0/1 (rows 16w..16w+15)
    int col  = blockIdx.x * 16 + (lane & 15);
    int kb   = (lane >> 4) * 2;             // K-pair within lane-half
    int arow = wv * 16 + (lane & 15);
    v8f acc = {};
#pragma unroll
    for (int k = 0; k < 16; ++k) {          // K=64 in steps of 4
        int kk = k * 4 + kb;
        v2f a = { pw_w[arow * 64 + kk], pw_w[arow * 64 + kk + 1] };
        v2f b = { g[(size_t)kk * NSP + col], g[(size_t)(kk + 1) * NSP + col] };
        acc = __builtin_amdgcn_wmma_f32_16x16x4_f32(false, a, false, b,
                                                    (short)0, acc, false, false);
    }
    int rbase = wv * 16 + ((lane >> 4) ? 8 : 0);
#pragma unroll
    for (int r = 0; r < 8; ++r) {
        int row = rbase + r;
        float v = acc[r] + pw_b[row];
        f[(size_t)row * NSP + col] = v > 0.f ? v : 0.f;
    }
}

// ---------------- Kernel 3: V = v_w(64x32) @ f via WMMA (n-major store), + kscore ----------------
__global__ __launch_bounds__(128) void gemm2_kernel(const float* __restrict__ f,
        const float* __restrict__ v_w, const float* __restrict__ k_w,
        float* __restrict__ vt, float* __restrict__ ks)
{
    __shared__ float tile[64 * 17];         // stride 17 -> bank-conflict-free transpose
    int lane = threadIdx.x & 31;
    int wv   = threadIdx.x >> 5;            // 0..3 -> V rows 16w..16w+15
    int colb = blockIdx.x * 16;
    int col  = colb + (lane & 15);
    int kb   = (lane >> 4) * 2;
    int arow = wv * 16 + (lane & 15);
    v8f acc = {};
#pragma unroll
    for (int k = 0; k < 8; ++k) {           // K=32 in steps of 4
        int kk = k * 4 + kb;
        v2f a = { v_w[arow * 32 + kk], v_w[arow * 32 + kk + 1] };
        v2f b = { f[(size_t)kk * NSP + col], f[(size_t)(kk + 1) * NSP + col] };
        acc = __builtin_amdgcn_wmma_f32_16x16x4_f32(false, a, false, b,
                                                    (short)0, acc, false, false);
    }
    int rbase = wv * 16 + ((lane >> 4) ? 8 : 0);
#pragma unroll
    for (int r = 0; r < 8; ++r)
        tile[(rbase + r) * 17 + (lane & 15)] = acc[r];

    // kscore: only K channel 0 of each head feeds the scores (rows 0, 32 of k_w)
    if (threadIdx.x < 32) {
        int h  = threadIdx.x >> 4;
        int cc = colb + (threadIdx.x & 15);
        const float* kr = k_w + h * 32 * 32;
        float s = 0.f;
#pragma unroll
        for (int c = 0; c < 32; ++c) s = fmaf(kr[c], f[(size_t)c * NSP + cc], s);
        ks[(size_t)h * NSP + cc] = s * 0.17677669529663687f;  // 1/sqrt(32)
    }
    __syncthreads();
#pragma unroll
    for (int i = 0; i < 8; ++i) {           // coalesced n-major store: Vt[n][64]
        int idx = threadIdx.x + i * 128;
        int c2  = idx >> 6;
        int ch  = idx & 63;
        vt[(size_t)(colb + c2) * 64 + ch] = tile[ch * 17 + c2];
    }
}

// ---------------- Kernel 4: windowed softmax attention + output head ----------------
__global__ __launch_bounds__(128) void attn_kernel(const float* __restrict__ ks,
        const float* __restrict__ vt, const float* __restrict__ out_w,
        const float* __restrict__ out_b, float* __restrict__ out)
{
    __shared__ int   sIdx[W3];
    __shared__ float sSc[2][W3];
    __shared__ float sRed[128];
    int g = blockIdx.x;
    int t = threadIdx.x;
    int a = g / 144, b = (g / 12) % 12, c = g % 12;
    // replicate numpy linspace/truncation windowing (double; frac parts >= .09)
    double cx = (double)(a * 47) / 11.0;
    double cy = (double)(b * 47) / 11.0;
    double cz = (double)(c * 47) / 11.0;
    int lox = (int)fmax(0.0, cx - 3.0), hix = (int)fmin(48.0, cx + 4.0);
    int loy = (int)fmax(0.0, cy - 3.0), hiy = (int)fmin(48.0, cy + 4.0);
    int loz = (int)fmax(0.0, cz - 3.0), hiz = (int)fmin(48.0, cz + 4.0);
    for (int w = t; w < W3; w += 128) {
        int wx = w / 49, wy = (w / 7) % 7, wz = w % 7;
        int ix = lox + wx, iy = loy + wy, iz = loz + wz;
        bool ok = (ix < hix) && (iy < hiy) && (iz < hiz);
        if (ok) {
            int n = (ix * 48 + iy) * 48 + iz;
            sIdx[w]   = n;
            sSc[0][w] = ks[n];
            sSc[1][w] = ks[NSP + n];
        } else {
            sIdx[w]   = -1;
            sSc[0][w] = -1e30f;
            sSc[1][w] = -1e30f;
        }
    }
    __syncthreads();
    int h = t >> 6, l = t & 63;
    // max
    float m = -1e30f;
    for (int w = l; w < W3; w += 64) m = fmaxf(m, sSc[h][w]);
    sRed[t] = m;
    __syncthreads();
    for (int s = 32; s > 0; s >>= 1) {
        if (l < s) sRed[t] = fmaxf(sRed[t], sRed[t + s]);
        __syncthreads();
    }
    float mx = sRed[h * 64];
    __syncthreads();
    // exp + sum
    float sum = 0.f;
    for (int w = l; w < W3; w += 64) {
        float e = __expf(sSc[h][w] - mx);
        sSc[h][w] = e;
        sum += e;
    }
    sRed[t] = sum;
    __syncthreads();
    for (int s = 32; s > 0; s >>= 1) {
        if (l < s) sRed[t] += sRed[t + s];
        __syncthreads();
    }
    float inv0 = 1.f / sRed[0];
    float inv1 = 1.f / sRed[64];
    __syncthreads();
    // ctx[h*32+c] = sum_w p * Vt[n][h*32+c]; 2-way split over w
    int part = t >> 6;
    int ch   = t & 63;
    float inv = (ch >> 5) ? inv1 : inv0;
    const float* scp = sSc[ch >> 5];
    float acc = 0.f;
    int w0 = part ? 172 : 0;
    int w1 = part ? W3 : 172;
    for (int w = w0; w < w1; ++w) {
        int n = sIdx[w];                    // uniform across lanes -> scalar branch
        if (n >= 0) acc = fmaf(scp[w] * inv, vt[(size_t)n * 64 + ch], acc);
    }
    sRed[t] = acc;
    __syncthreads();
    if (t < 3) {
        float o = out_b[t];
        for (int cc2 = 0; cc2 < 64; ++cc2)
            o = fmaf(out_w[t * 64 + cc2], sRed[cc2] + sRed[64 + cc2], o);
        out[t * G_CNT + g] = o;
    }
}

extern "C" void kernel_launch(void* const* d_in, const int* in_sizes, int n_in,
                              void* d_out, int out_size, void* d_ws, size_t ws_size,
                              hipStream_t stream)
{
    const float* feat  = (const float*)d_in[0];
    const float* dw_w  = (const float*)d_in[1];
    const float* dw_b  = (const float*)d_in[2];
    const float* pw_w  = (const float*)d_in[3];
    const float* pw_b  = (const float*)d_in[4];
    const float* k_w   = (const float*)d_in[5];
    const float* v_w   = (const float*)d_in[6];
    const float* out_w = (const float*)d_in[7];
    const float* out_b = (const float*)d_in[8];
    float* out = (float*)d_out;

    float* ws   = (float*)d_ws;
    float* gbuf = ws;                        // 64*NSP floats, reused as Vt[n][64]
    float* fbuf = ws + (size_t)64 * NSP;     // 32*NSP floats
    float* ksb  = fbuf + (size_t)32 * NSP;   // 2*NSP floats

    dwconv_kernel<<<(64 * NSP) / 256, 256, 0, stream>>>(feat, dw_w, dw_b, gbuf);
    gemm1_kernel<<<NSP / 16, 64, 0, stream>>>(gbuf, pw_w, pw_b, fbuf);
    gemm2_kernel<<<NSP / 16, 128, 0, stream>>>(fbuf, v_w, k_w, gbuf, ksb);
    attn_kernel<<<G_CNT, 128, 0, stream>>>(ksb, gbuf, out_w, out_b, out);
}